// SampleConv_481036337461
// MI455X (gfx1250) — compile-verified
//
#include <hip/hip_runtime.h>

// SampleConv for MI455X (gfx1250):
//   h = x @ W^T + b        via V_WMMA_F32_16X16X4_F32 (full fp32 tensor path)
//   out = scatter_mean(h[src] -> dst)  via hw global_atomic_add_f32 (L2-resident)
//
// Sizes fixed by the reference:
#define N_NODES 50000
#define E_EDGES 800000
#define DIN     64
#define DOUT    64

typedef __attribute__((ext_vector_type(2))) float v2f;
typedef __attribute__((ext_vector_type(8))) float v8f;

// ---------------------------------------------------------------------------
// Zero the accumulation buffers (harness poisons them with 0xAA).
// ---------------------------------------------------------------------------
__global__ void zero_kernel(float* __restrict__ out, float* __restrict__ cnt) {
  int i = blockIdx.x * blockDim.x + threadIdx.x;
  if (i < N_NODES * DOUT) out[i] = 0.0f;
  if (i < N_NODES)        cnt[i] = 0.0f;
}

// ---------------------------------------------------------------------------
// h[m][n] = sum_k x[m][k] * W[n][k] + b[n]
// One wave32 computes a 16(M)x16(N) tile; 4 waves per block cover DOUT=64.
// A-matrix (16x4 f32, 2 VGPRs): lane l<16 holds row M=l, K={k0,k0+1};
//                               lane l>=16 holds row M=l-16, K={k0+2,k0+3}.
// B-matrix (4x16 f32, 2 VGPRs): mirrored — lane half selects the K-pair,
//                               lane%16 selects column N. B[k][n] = W[n][k].
// C/D (16x16 f32, 8 VGPRs): VGPR r, lane l: M = r + 8*(l>=16), N = l%16.
// ---------------------------------------------------------------------------
__global__ __launch_bounds__(128) void linear_wmma_kernel(
    const float* __restrict__ x, const float* __restrict__ W,
    const float* __restrict__ b, float* __restrict__ h) {
  const int lane  = threadIdx.x & 31;
  const int wave  = threadIdx.x >> 5;     // 0..3 -> N tile within DOUT
  const int mBase = blockIdx.x * 16;      // 3125 blocks * 16 = 50000 exactly
  const int nBase = wave * 16;
  const int half  = lane >> 4;            // 0: K pair {0,1}; 1: K pair {2,3}
  const int l     = lane & 15;

  v8f acc = {};
  const float* ap = x + (size_t)(mBase + l) * DIN;  // A row (x)
  const float* bp = W + (size_t)(nBase + l) * DIN;  // B "column" (row of W)

#pragma unroll
  for (int k0 = 0; k0 < DIN; k0 += 4) {
    const int ka = k0 + half * 2;
    v2f a;  a.x = ap[ka]; a.y = ap[ka + 1];
    v2f bb; bb.x = bp[ka]; bb.y = bp[ka + 1];
    // 8 args: (neg_a, A, neg_b, B, c_mod, C, reuse_a, reuse_b)
    acc = __builtin_amdgcn_wmma_f32_16x16x4_f32(
        false, a, false, bb, (short)0, acc, false, false);
  }

  const int col = nBase + l;
  const float bias = b[col];
#pragma unroll
  for (int r = 0; r < 8; ++r) {
    const int row = mBase + r + half * 8;
    h[(size_t)row * DOUT + col] = acc[r] + bias;
  }
}

// ---------------------------------------------------------------------------
// Edge scatter: 16 threads per edge, float4 per thread.
// h (12.8 MB) and out (12.8 MB) are L2-resident on the 192 MB L2, so the
// gather is an L2 read and the atomic add resolves in the L2 atomic units.
// unsafeAtomicAdd -> global_atomic_add_f32 (no-return, STOREcnt path).
// ---------------------------------------------------------------------------
__global__ void scatter_kernel(const int* __restrict__ src,
                               const int* __restrict__ dst,
                               const float* __restrict__ h,
                               float* __restrict__ out,
                               float* __restrict__ cnt) {
  const long long t = (long long)blockIdx.x * blockDim.x + threadIdx.x;
  if (t >= (long long)E_EDGES * 16) return;
  const int e = (int)(t >> 4);
  const int f = ((int)t & 15) * 4;
  const int s = src[e];
  const int d = dst[e];
  const float4 v = *(const float4*)(h + (size_t)s * DOUT + f);
  float* o = out + (size_t)d * DOUT + f;
  unsafeAtomicAdd(o + 0, v.x);
  unsafeAtomicAdd(o + 1, v.y);
  unsafeAtomicAdd(o + 2, v.z);
  unsafeAtomicAdd(o + 3, v.w);
  if (((int)t & 15) == 0) unsafeAtomicAdd(cnt + d, 1.0f);
}

// ---------------------------------------------------------------------------
// out[node][:] /= max(cnt[node], 1)
// ---------------------------------------------------------------------------
__global__ void finalize_kernel(float* __restrict__ out,
                                const float* __restrict__ cnt) {
  const int i = blockIdx.x * blockDim.x + threadIdx.x;
  if (i >= N_NODES * 16) return;
  const int node = i >> 4;
  const int f = (i & 15) * 4;
  const float inv = 1.0f / fmaxf(cnt[node], 1.0f);
  float4* p = (float4*)(out + (size_t)node * DOUT + f);
  float4 v = *p;
  v.x *= inv; v.y *= inv; v.z *= inv; v.w *= inv;
  *p = v;
}

extern "C" void kernel_launch(void* const* d_in, const int* in_sizes, int n_in,
                              void* d_out, int out_size, void* d_ws, size_t ws_size,
                              hipStream_t stream) {
  const float* x  = (const float*)d_in[0];              // [N, DIN]
  const int*   ei = (const int*)d_in[1];                // [2, E] row0=src, row1=dst
  const float* W  = (const float*)d_in[2];              // [DOUT, DIN]
  const float* b  = (const float*)d_in[3];              // [DOUT]
  float* out = (float*)d_out;                           // [N, DOUT]

  // Workspace: h [N*DOUT] floats, then cnt [N] floats.
  float* h   = (float*)d_ws;
  float* cnt = (float*)((char*)d_ws + (size_t)N_NODES * DOUT * sizeof(float));

  // 1) zero accumulators
  {
    const int total = N_NODES * DOUT;
    zero_kernel<<<(total + 255) / 256, 256, 0, stream>>>(out, cnt);
  }
  // 2) h = x W^T + b   (WMMA fp32)
  linear_wmma_kernel<<<N_NODES / 16, 128, 0, stream>>>(x, W, b, h);
  // 3) scatter-add h[src] -> out[dst], count edges per dst
  {
    const long long total = (long long)E_EDGES * 16;
    scatter_kernel<<<(int)((total + 255) / 256), 256, 0, stream>>>(
        ei, ei + E_EDGES, h, out, cnt);
  }
  // 4) divide by degree (min 1)
  finalize_kernel<<<(N_NODES * 16 + 255) / 256, 256, 0, stream>>>(out, cnt);
}